// VLFAN_90941637525533
// MI455X (gfx1250) — compile-verified
//
#include <hip/hip_runtime.h>
#include <math.h>

typedef __attribute__((ext_vector_type(2))) float v2f;
typedef __attribute__((ext_vector_type(8))) float v8f;

#define C_DIM 1024
#define NQ 10
#define N_ROWS 100000
#define TILES (N_ROWS / 16)
#define EPSV 1e-12f
#define LSCALE 100.0f

// workspace layout (float offsets)
#define WS_QN 0                          // 16*1024 padded normalized queries
#define WS_LOGIT 16384                   // N_ROWS*10 logits
#define WS_MAX (WS_LOGIT + N_ROWS * NQ)  // 16 flipped-uint per-query maxima
#define WS_SUM (WS_MAX + 16)             // 16 per-query exp-sums
#define WS_OUT (WS_SUM + 16)             // 16*1024 unnormalized out accum
#define WS_POOL (WS_OUT + 16384)         // 1024 pooled

__device__ __forceinline__ unsigned flipf(float f) {
  unsigned u = __float_as_uint(f);
  return (u & 0x80000000u) ? ~u : (u | 0x80000000u);
}
__device__ __forceinline__ float unflipf(unsigned u) {
  return __uint_as_float((u & 0x80000000u) ? (u ^ 0x80000000u) : ~u);
}

// ---------------- init: zero accumulators / maxima ----------------
__global__ void k_init(float* ws) {
  int i = blockIdx.x * blockDim.x + threadIdx.x;
  if (i < 16 * C_DIM) ws[WS_OUT + i] = 0.0f;
  if (i < 16) {
    ((unsigned*)(ws + WS_MAX))[i] = 0u;  // flipped -inf sentinel
    ws[WS_SUM + i] = 0.0f;
  }
}

// ---------------- normalize queries into padded 16x1024 ----------------
__global__ void k_qnorm(const float* __restrict__ Q, float* __restrict__ ws) {
  int b = blockIdx.x, t = threadIdx.x;
  float* Qn = ws + WS_QN;
  if (b >= NQ + 1) {
    for (int i = t; i < C_DIM; i += 256) Qn[b * C_DIM + i] = 0.0f;
    return;
  }
  float s = 0.0f;
  for (int i = t; i < C_DIM; i += 256) {
    float v = Q[b * C_DIM + i];
    s = fmaf(v, v, s);
  }
  for (int m = 16; m >= 1; m >>= 1) s += __shfl_xor(s, m, 32);
  __shared__ float sm[8];
  int lane = t & 31, wv = t >> 5;
  if (lane == 0) sm[wv] = s;
  __syncthreads();
  if (t == 0) {
    float tot = 0.0f;
    for (int i = 0; i < 8; i++) tot += sm[i];
    sm[0] = 1.0f / fmaxf(sqrtf(tot), EPSV);
  }
  __syncthreads();
  float rn = sm[0];
  for (int i = t; i < C_DIM; i += 256) Qn[b * C_DIM + i] = Q[b * C_DIM + i] * rn;
}

// ---------------- pass 1: logits = 100*(cos(q,x) - cos(q10,x)) via WMMA ----
__global__ void k_scores(const float* __restrict__ X, float* __restrict__ ws) {
  const float* Qn = ws + WS_QN;
  float* Lw = ws + WS_LOGIT;
  unsigned* maxw = (unsigned*)(ws + WS_MAX);
  int lane = threadIdx.x & 31;
  int wave = threadIdx.x >> 5;
  int gw = blockIdx.x * (blockDim.x >> 5) + wave;
  int nwaves = gridDim.x * (blockDim.x >> 5);
  int hi = lane >> 4;    // half-wave selector
  int m = lane & 15;     // A row (query) == B column (X row in tile)
  for (int t = gw; t < TILES; t += nwaves) {
    int row0 = t * 16;
    int r = row0 + m;
    const float* xr = X + (size_t)r * C_DIM + 2 * hi;
    const float* qr = Qn + m * C_DIM + 2 * hi;
    v8f c = {0.f, 0.f, 0.f, 0.f, 0.f, 0.f, 0.f, 0.f};
    float ssq = 0.0f;
    for (int k = 0; k < C_DIM; k += 4) {
      v2f bv = *(const v2f*)(xr + k);   // B[k+2hi..+1][m] = X[r][k+2hi..+1]
      v2f av = *(const v2f*)(qr + k);   // A[m][k+2hi..+1] = Qn[m][k+2hi..+1]
      ssq = fmaf(bv.x, bv.x, fmaf(bv.y, bv.y, ssq));
      c = __builtin_amdgcn_wmma_f32_16x16x4_f32(false, av, false, bv,
                                                (short)0, c, false, false);
    }
    ssq += __shfl_xor(ssq, 16, 32);  // full ||x||^2 for row m
    float rn = 1.0f / fmaxf(sqrtf(ssq), EPSV);
    // gated query m=10 lives in VGPR 2 of the upper half-lanes
    float g = __shfl_xor(c[2], 16, 32);
    float q10 = hi ? c[2] : g;
    float l[8];
#pragma unroll
    for (int j = 0; j < 8; j++) l[j] = LSCALE * rn * (c[j] - q10);
    if (hi == 0) {
#pragma unroll
      for (int j = 0; j < 8; j++) Lw[(size_t)r * NQ + j] = l[j];  // q=0..7
    } else {
      Lw[(size_t)r * NQ + 8] = l[0];  // q=8
      Lw[(size_t)r * NQ + 9] = l[1];  // q=9
    }
    // max over the 16 rows of the tile (within each half-wave)
#pragma unroll
    for (int j = 0; j < 8; j++) {
      float v = l[j];
      v = fmaxf(v, __shfl_xor(v, 1, 32));
      v = fmaxf(v, __shfl_xor(v, 2, 32));
      v = fmaxf(v, __shfl_xor(v, 4, 32));
      v = fmaxf(v, __shfl_xor(v, 8, 32));
      l[j] = v;
    }
    if (lane == 0) {
#pragma unroll
      for (int j = 0; j < 8; j++) atomicMax(&maxw[j], flipf(l[j]));
    } else if (lane == 16) {
      atomicMax(&maxw[8], flipf(l[0]));
      atomicMax(&maxw[9], flipf(l[1]));
    }
  }
}

// ---------------- pass 2: out[q][c] = sum_n exp(l-max) * X[n][c] via WMMA ---
#define CG 2                               // channel groups (32 waves each)
#define NCHUNK 250                         // row chunks
#define ROWS_PER_CHUNK (N_ROWS / NCHUNK)   // 400
#define RB 64                              // rows per LDS weight batch

__global__ __launch_bounds__(1024) void k_attnmm(const float* __restrict__ X,
                                                 float* __restrict__ ws) {
  const float* Lw = ws + WS_LOGIT;
  const unsigned* maxw = (const unsigned*)(ws + WS_MAX);
  float* sums = ws + WS_SUM;
  float* outacc = ws + WS_OUT;
  __shared__ float sw[RB * 16];
  __shared__ float smax[16];
  __shared__ float ssum[16];
  int tid = threadIdx.x;
  int cg = blockIdx.x % CG;
  int chunk = blockIdx.x / CG;
  int lane = tid & 31, wave = tid >> 5;
  int hi = lane >> 4, m = lane & 15;
  int c0 = (cg * 32 + wave) * 16;  // this wave's 16-channel tile
  if (tid < 16) {
    smax[tid] = (tid < NQ) ? unflipf(maxw[tid]) : 0.0f;
    ssum[tid] = 0.0f;
  }
  int rbase = chunk * ROWS_PER_CHUNK;
  int rend = rbase + ROWS_PER_CHUNK;
  int wrow = tid >> 4;  // 0..63 row-in-batch
  int wq = tid & 15;    // 0..15 query slot
  float sumLocal = 0.0f;
  v8f c = {0.f, 0.f, 0.f, 0.f, 0.f, 0.f, 0.f, 0.f};
  for (int rb = rbase; rb < rend; rb += RB) {
    int brows = min(RB, rend - rb);  // multiple of 4 by construction
    __syncthreads();                 // previous batch readers done
    float wv = 0.0f;
    if (wrow < brows && wq < NQ) {
      float lg = Lw[(size_t)(rb + wrow) * NQ + wq];
      wv = __expf(lg - smax[wq]);
      if (cg == 0) sumLocal += wv;
    }
    sw[wrow * 16 + wq] = wv;
    __syncthreads();
    for (int kk = 0; kk < brows; kk += 4) {
      int r = rb + kk + 2 * hi;
      v2f bv;  // B[k][n] = X[row k][c0+n]
      bv.x = X[(size_t)r * C_DIM + c0 + m];
      bv.y = X[(size_t)(r + 1) * C_DIM + c0 + m];
      v2f av;  // A[m][k] = w_m(row k)
      av.x = sw[(kk + 2 * hi) * 16 + m];
      av.y = sw[(kk + 2 * hi + 1) * 16 + m];
      c = __builtin_amdgcn_wmma_f32_16x16x4_f32(false, av, false, bv,
                                                (short)0, c, false, false);
    }
  }
#pragma unroll
  for (int j = 0; j < 8; j++) {
    int q = j + 8 * hi;
    atomicAdd(&outacc[q * C_DIM + c0 + m], c[j]);
  }
  if (cg == 0) {  // block-uniform branch: barriers safe
    if (wq < NQ) atomicAdd(&ssum[wq], sumLocal);
    __syncthreads();
    if (tid < NQ) atomicAdd(&sums[tid], ssum[tid]);
  }
}

// ---------------- finalize: normalize, mean-pool ----------------
__global__ void k_pool(float* __restrict__ ws) {
  int i = blockIdx.x * 256 + threadIdx.x;  // 0..1023
  const float* outacc = ws + WS_OUT;
  const float* sums = ws + WS_SUM;
  float acc = 0.0f;
  for (int q = 0; q < NQ; q++) acc += outacc[q * C_DIM + i] / sums[q];
  ws[WS_POOL + i] = acc * (1.0f / NQ);
}

// ---------------- adapter GEMV: out[i] = pooled . W[i,:] + b[i] ------------
__global__ void k_adapter(const float* __restrict__ W,
                          const float* __restrict__ bvec,
                          const float* __restrict__ ws, float* __restrict__ out) {
  int rowi = blockIdx.x;
  const float* pooled = ws + WS_POOL;
  float s = 0.0f;
  for (int ci = threadIdx.x; ci < C_DIM; ci += 256)
    s = fmaf(pooled[ci], W[(size_t)rowi * C_DIM + ci], s);
  for (int msk = 16; msk >= 1; msk >>= 1) s += __shfl_xor(s, msk, 32);
  __shared__ float sm[8];
  int lane = threadIdx.x & 31, wv = threadIdx.x >> 5;
  if (lane == 0) sm[wv] = s;
  __syncthreads();
  if (threadIdx.x == 0) {
    float tot = 0.0f;
    for (int i = 0; i < 8; i++) tot += sm[i];
    out[rowi] = tot + bvec[rowi];
  }
}

extern "C" void kernel_launch(void* const* d_in, const int* in_sizes, int n_in,
                              void* d_out, int out_size, void* d_ws,
                              size_t ws_size, hipStream_t stream) {
  (void)in_sizes; (void)n_in; (void)out_size; (void)ws_size;
  const float* X = (const float*)d_in[0];
  const float* Q = (const float*)d_in[1];
  const float* W = (const float*)d_in[2];
  const float* b = (const float*)d_in[3];
  float* out = (float*)d_out;
  float* ws = (float*)d_ws;

  k_init<<<64, 256, 0, stream>>>(ws);
  k_qnorm<<<NQ + 6, 256, 0, stream>>>(Q, ws);       // 16 blocks (padded rows)
  k_scores<<<782, 256, 0, stream>>>(X, ws);         // 6256 waves for 6250 tiles
  k_attnmm<<<CG * NCHUNK, 1024, 0, stream>>>(X, ws);
  k_pool<<<4, 256, 0, stream>>>(ws);
  k_adapter<<<C_DIM, 256, 0, stream>>>(W, b, ws, out);
}